// WeaveLayer_14705968022036
// MI455X (gfx1250) — compile-verified
//
#include <hip/hip_runtime.h>

typedef __attribute__((ext_vector_type(2))) float v2f;
typedef __attribute__((ext_vector_type(8))) float v8f;

#define N_ATOMS 50000
#define N_PAIRS 500000
#define A_IN 75
#define P_IN 14
#define H_A 100
#define H_P 50
#define A_OUT 50
#define P_OUT 50

#define XLD 153   // padded row stride for concat(AA,PA) tile in LDS (odd -> no bank conflicts)
#define HLD 105   // padded row stride for pair hidden tile in LDS

__device__ __forceinline__ v8f wmma_f32(v2f a, v2f b, v8f c) {
  // D = A(16x4,f32) * B(4x16,f32) + C(16x16,f32)
  return __builtin_amdgcn_wmma_f32_16x16x4_f32(false, a, false, b, (short)0, c,
                                               false, false);
}

__device__ __forceinline__ float frelu(float x) { return x > 0.0f ? x : 0.0f; }

// stage a [K][N] row-major weight matrix into LDS, zero-padded to [KP][NP]
__device__ __forceinline__ void stage_w(const float* __restrict__ W, float* Ws,
                                        int K, int N, int KP, int NP, int tid,
                                        int nth) {
  for (int i = tid; i < KP * NP; i += nth) {
    int k = i / NP, n = i - k * NP;
    Ws[i] = (k < K && n < N) ? W[k * N + n] : 0.0f;
  }
}
__device__ __forceinline__ void stage_b(const float* __restrict__ b, float* bs,
                                        int N, int NP, int tid, int nth) {
  for (int i = tid; i < NP; i += nth) bs[i] = (i < N) ? b[i] : 0.0f;
}

// ---------------------------------------------------------------------------
// Kernel 1: deterministic segment-sum  PA_sum[a,:] = sum_{p: split[p]==a}
//           relu(pair_features[p] @ W_pa + b_pa)
// pair_split is sorted, so each atom's pairs form a contiguous range found by
// binary search. One block (64 threads) per atom; thread c owns column c.
// ---------------------------------------------------------------------------
__global__ __launch_bounds__(64) void pa_segsum_det(
    const float* __restrict__ pf, const int* __restrict__ pair_split,
    const float* __restrict__ W_pa, const float* __restrict__ b_pa,
    float* __restrict__ PA_sum) {
  __shared__ float Ws[P_IN * H_P];
  __shared__ float bs[H_P];
  const int tid = threadIdx.x;
  for (int i = tid; i < P_IN * H_P; i += blockDim.x) Ws[i] = W_pa[i];
  for (int i = tid; i < H_P; i += blockDim.x) bs[i] = b_pa[i];
  __syncthreads();

  const int a = blockIdx.x;
  // lower_bound(a) .. lower_bound(a+1)
  int lo = 0, hi = N_PAIRS;
  while (lo < hi) {
    int mid = (lo + hi) >> 1;
    if (pair_split[mid] < a) lo = mid + 1; else hi = mid;
  }
  int lo2 = lo, hi2 = N_PAIRS;
  while (lo2 < hi2) {
    int mid = (lo2 + hi2) >> 1;
    if (pair_split[mid] < a + 1) lo2 = mid + 1; else hi2 = mid;
  }

  const int c = tid;
  if (c < H_P) {
    float acc = 0.0f;
    for (int p = lo; p < lo2; ++p) {
      const float* x = pf + (size_t)p * P_IN;
      float v = bs[c];
#pragma unroll
      for (int k = 0; k < P_IN; ++k) v += x[k] * Ws[k * H_P + c];
      acc += frelu(v);
    }
    PA_sum[(size_t)a * H_P + c] = acc;
  }
}

// ---------------------------------------------------------------------------
// Kernel 2: fused atom branch.
//   AA = relu(atom @ W_aa + b_aa)            (WMMA, K=75->76)
//   X  = concat(AA, PA_sum)  (LDS, K padded 150->152)
//   A  = relu(X @ W_ao + b_ao)               (WMMA, K=152)
// 4 waves / block, one 16-atom row tile per wave.
// ---------------------------------------------------------------------------
#define K3_WAVES 4
__global__ __launch_bounds__(128) void atom_kernel(
    const float* __restrict__ atomf, const float* __restrict__ PA_sum,
    const float* __restrict__ W_aa, const float* __restrict__ b_aa,
    const float* __restrict__ W_ao, const float* __restrict__ b_ao,
    float* __restrict__ outA) {
  __shared__ float Waa[76 * 112];
  __shared__ float baa[112];
  __shared__ float Wao[152 * 64];
  __shared__ float bao[64];
  __shared__ float Xs[K3_WAVES][16 * XLD];

  const int tid = threadIdx.x, nth = blockDim.x;
  stage_w(W_aa, Waa, A_IN, H_A, 76, 112, tid, nth);
  stage_b(b_aa, baa, H_A, 112, tid, nth);
  stage_w(W_ao, Wao, H_A + H_P, A_OUT, 152, 64, tid, nth);
  stage_b(b_ao, bao, A_OUT, 64, tid, nth);
  __syncthreads();

  const int wv = tid >> 5, lane = tid & 31;
  const int half = lane >> 4, lm = lane & 15, koff = half * 2;
  const int nTiles = N_ATOMS / 16;
  const int tileRaw = blockIdx.x * K3_WAVES + wv;
  const int tile = tileRaw < nTiles ? tileRaw : nTiles - 1;
  const bool valid = tileRaw < nTiles;
  const int a0 = tile * 16;

  float* X = &Xs[wv][0];
  const float* Arow = atomf + (size_t)(a0 + lm) * A_IN;

  // ---- phase 1: AA tile into X[:, 0:100] ----
  for (int ct = 0; ct < 7; ++ct) {
    v8f acc = {0.f, 0.f, 0.f, 0.f, 0.f, 0.f, 0.f, 0.f};
    for (int k0 = 0; k0 < 76; k0 += 4) {
      int k = k0 + koff;
      v2f af;
      af.x = (k < A_IN) ? Arow[k] : 0.0f;
      af.y = (k + 1 < A_IN) ? Arow[k + 1] : 0.0f;
      v2f bf;
      bf.x = Waa[k * 112 + ct * 16 + lm];
      bf.y = Waa[(k + 1) * 112 + ct * 16 + lm];
      acc = wmma_f32(af, bf, acc);
    }
    int col = ct * 16 + lm;
    if (col < H_A) {
      float bb = baa[col];
#pragma unroll
      for (int r = 0; r < 8; ++r)
        X[(r + half * 8) * XLD + col] = frelu(acc[r] + bb);
    }
  }
  // ---- X[:, 100:150] = PA_sum rows; X[:, 150:152] = 0 ----
  for (int i = lane; i < 16 * 52; i += 32) {
    int r = i / 52, c = i - r * 52;
    X[r * XLD + 100 + c] =
        (c < H_P) ? PA_sum[(size_t)(a0 + r) * H_P + c] : 0.0f;
  }
  __syncthreads();

  // ---- phase 2: A = relu(X @ W_ao + b_ao) ----
  for (int ct = 0; ct < 4; ++ct) {
    v8f acc = {0.f, 0.f, 0.f, 0.f, 0.f, 0.f, 0.f, 0.f};
    for (int k0 = 0; k0 < 152; k0 += 4) {
      int k = k0 + koff;
      v2f af;
      af.x = X[lm * XLD + k];
      af.y = X[lm * XLD + k + 1];
      v2f bf;
      bf.x = Wao[k * 64 + ct * 16 + lm];
      bf.y = Wao[(k + 1) * 64 + ct * 16 + lm];
      acc = wmma_f32(af, bf, acc);
    }
    int col = ct * 16 + lm;
    if (valid && col < A_OUT) {
      float bb = bao[col];
#pragma unroll
      for (int r = 0; r < 8; ++r)
        outA[(size_t)(a0 + r + half * 8) * A_OUT + col] = frelu(acc[r] + bb);
    }
  }
}

// ---------------------------------------------------------------------------
// Kernel 3: pair branch.
//   AP_ij = relu([a_i;a_j] @ W_ap + b)   AP_ji = relu([a_j;a_i] @ W_ap + b)
//   PP    = relu(pair @ W_pp + b)
//   H     = concat(AP_ij + AP_ji, PP)    (LDS)
//   P     = relu(H @ W_po + b_po)
// 8 waves / block, one 16-pair row tile per wave. The ji input is the column
// swap of the ij input, so each k-step loads a_i[k]/a_j[k] once and feeds
// both accumulator sets.
// ---------------------------------------------------------------------------
#define K4_WAVES 8
__global__ __launch_bounds__(256) void pair_kernel(
    const float* __restrict__ atomf, const float* __restrict__ pf,
    const int* __restrict__ a2p, const float* __restrict__ W_ap,
    const float* __restrict__ b_ap, const float* __restrict__ W_pp,
    const float* __restrict__ b_pp, const float* __restrict__ W_po,
    const float* __restrict__ b_po, float* __restrict__ outP) {
  __shared__ float Wap[152 * 64];
  __shared__ float bap[64];
  __shared__ float Wpp[16 * 64];
  __shared__ float bpp[64];
  __shared__ float Wpo[104 * 64];
  __shared__ float bpo[64];
  __shared__ float Hs[K4_WAVES][16 * HLD];

  const int tid = threadIdx.x, nth = blockDim.x;
  stage_w(W_ap, Wap, 2 * A_IN, H_P, 152, 64, tid, nth);
  stage_b(b_ap, bap, H_P, 64, tid, nth);
  stage_w(W_pp, Wpp, P_IN, H_P, 16, 64, tid, nth);
  stage_b(b_pp, bpp, H_P, 64, tid, nth);
  stage_w(W_po, Wpo, 2 * H_P, P_OUT, 104, 64, tid, nth);
  stage_b(b_po, bpo, P_OUT, 64, tid, nth);
  __syncthreads();

  const int wv = tid >> 5, lane = tid & 31;
  const int half = lane >> 4, lm = lane & 15, koff = half * 2;
  const int nTiles = N_PAIRS / 16;
  const int tileRaw = blockIdx.x * K4_WAVES + wv;
  const int tile = tileRaw < nTiles ? tileRaw : nTiles - 1;
  const bool valid = tileRaw < nTiles;
  const int p0 = tile * 16;
  float* H = &Hs[wv][0];

  const int ii = a2p[2 * (p0 + lm) + 0];
  const int jj = a2p[2 * (p0 + lm) + 1];
  const float* Ai = atomf + (size_t)ii * A_IN;
  const float* Aj = atomf + (size_t)jj * A_IN;

  // ---- AP_ij / AP_ji, all 4 col tiles in one K pass ----
  v8f aij[4], aji[4];
  {
    v8f z = {0.f, 0.f, 0.f, 0.f, 0.f, 0.f, 0.f, 0.f};
#pragma unroll
    for (int ct = 0; ct < 4; ++ct) { aij[ct] = z; aji[ct] = z; }
  }
  for (int k0 = 0; k0 < 152; k0 += 4) {
    int k = k0 + koff;
    v2f fij, fji;
    if (k < A_IN) {
      fij.x = Ai[k];
      fji.x = Aj[k];
    } else if (k < 2 * A_IN) {
      fij.x = Aj[k - A_IN];
      fji.x = Ai[k - A_IN];
    } else {
      fij.x = 0.0f;
      fji.x = 0.0f;
    }
    int k1 = k + 1;
    if (k1 < A_IN) {
      fij.y = Ai[k1];
      fji.y = Aj[k1];
    } else if (k1 < 2 * A_IN) {
      fij.y = Aj[k1 - A_IN];
      fji.y = Ai[k1 - A_IN];
    } else {
      fij.y = 0.0f;
      fji.y = 0.0f;
    }
#pragma unroll
    for (int ct = 0; ct < 4; ++ct) {
      v2f bf;
      bf.x = Wap[k * 64 + ct * 16 + lm];
      bf.y = Wap[(k + 1) * 64 + ct * 16 + lm];
      aij[ct] = wmma_f32(fij, bf, aij[ct]);
      aji[ct] = wmma_f32(fji, bf, aji[ct]);
    }
  }
#pragma unroll
  for (int ct = 0; ct < 4; ++ct) {
    int col = ct * 16 + lm;
    if (col < H_P) {
      float bb = bap[col];
#pragma unroll
      for (int r = 0; r < 8; ++r)
        H[(r + half * 8) * HLD + col] =
            frelu(aij[ct][r] + bb) + frelu(aji[ct][r] + bb);
    }
  }

  // ---- PP -> H[:, 50:100] ----
  const float* Prow = pf + (size_t)(p0 + lm) * P_IN;
  for (int ct = 0; ct < 4; ++ct) {
    v8f acc = {0.f, 0.f, 0.f, 0.f, 0.f, 0.f, 0.f, 0.f};
#pragma unroll
    for (int k0 = 0; k0 < 16; k0 += 4) {
      int k = k0 + koff;
      v2f af;
      af.x = (k < P_IN) ? Prow[k] : 0.0f;
      af.y = (k + 1 < P_IN) ? Prow[k + 1] : 0.0f;
      v2f bf;
      bf.x = Wpp[k * 64 + ct * 16 + lm];
      bf.y = Wpp[(k + 1) * 64 + ct * 16 + lm];
      acc = wmma_f32(af, bf, acc);
    }
    int col = ct * 16 + lm;
    if (col < H_P) {
      float bb = bpp[col];
#pragma unroll
      for (int r = 0; r < 8; ++r)
        H[(r + half * 8) * HLD + H_P + col] = frelu(acc[r] + bb);
    }
  }
  // zero K-pad columns 100..103
  for (int i = lane; i < 16 * 4; i += 32) {
    int r = i >> 2, c = i & 3;
    H[r * HLD + 100 + c] = 0.0f;
  }
  __syncthreads();

  // ---- P = relu(H @ W_po + b_po) ----
  for (int ct = 0; ct < 4; ++ct) {
    v8f acc = {0.f, 0.f, 0.f, 0.f, 0.f, 0.f, 0.f, 0.f};
    for (int k0 = 0; k0 < 104; k0 += 4) {
      int k = k0 + koff;
      v2f af;
      af.x = H[lm * HLD + k];
      af.y = H[lm * HLD + k + 1];
      v2f bf;
      bf.x = Wpo[k * 64 + ct * 16 + lm];
      bf.y = Wpo[(k + 1) * 64 + ct * 16 + lm];
      acc = wmma_f32(af, bf, acc);
    }
    int col = ct * 16 + lm;
    if (valid && col < P_OUT) {
      float bb = bpo[col];
#pragma unroll
      for (int r = 0; r < 8; ++r)
        outP[(size_t)(p0 + r + half * 8) * P_OUT + col] = frelu(acc[r] + bb);
    }
  }
}

// ---------------------------------------------------------------------------
extern "C" void kernel_launch(void* const* d_in, const int* in_sizes, int n_in,
                              void* d_out, int out_size, void* d_ws,
                              size_t ws_size, hipStream_t stream) {
  (void)in_sizes; (void)n_in; (void)out_size; (void)ws_size;
  const float* atom_features = (const float*)d_in[0];
  const float* pair_features = (const float*)d_in[1];
  const int* pair_split = (const int*)d_in[2];
  const int* atom_to_pair = (const int*)d_in[3];
  const float* W_aa = (const float*)d_in[4];
  const float* b_aa = (const float*)d_in[5];
  const float* W_pa = (const float*)d_in[6];
  const float* b_pa = (const float*)d_in[7];
  const float* W_ao = (const float*)d_in[8];
  const float* b_ao = (const float*)d_in[9];
  const float* W_ap = (const float*)d_in[10];
  const float* b_ap = (const float*)d_in[11];
  const float* W_pp = (const float*)d_in[12];
  const float* b_pp = (const float*)d_in[13];
  const float* W_po = (const float*)d_in[14];
  const float* b_po = (const float*)d_in[15];

  float* outA = (float*)d_out;
  float* outP = outA + (size_t)N_ATOMS * A_OUT;
  float* PA_sum = (float*)d_ws;  // [N_ATOMS, H_P] = 10 MB scratch

  // 1) deterministic segment-sum of PA
  pa_segsum_det<<<N_ATOMS, 64, 0, stream>>>(pair_features, pair_split, W_pa,
                                            b_pa, PA_sum);

  // 2) fused atom branch -> A
  const int atomTiles = N_ATOMS / 16;
  atom_kernel<<<(atomTiles + K3_WAVES - 1) / K3_WAVES, 32 * K3_WAVES, 0,
                stream>>>(atom_features, PA_sum, W_aa, b_aa, W_ao, b_ao, outA);

  // 3) pair branch -> P
  const int pairTiles = N_PAIRS / 16;
  pair_kernel<<<(pairTiles + K4_WAVES - 1) / K4_WAVES, 32 * K4_WAVES, 0,
                stream>>>(atom_features, pair_features, atom_to_pair, W_ap,
                          b_ap, W_pp, b_pp, W_po, b_po, outP);
}